// DCRNNet_16338055594466
// MI455X (gfx1250) — compile-verified
//
#include <hip/hip_runtime.h>
#include <hip/hip_bf16.h>
#include <math.h>

typedef float v2f __attribute__((ext_vector_type(2)));
typedef float v8f __attribute__((ext_vector_type(8)));

#define NN 100000
#define NE 1600000
// IN_C = 32, OUT_C = 64, OUT_SIZE = 32, K = 3, cin = 96 (only first 32 rows used)

// ---------------------------------------------------------------------------
// Zero workspace (harness poisons ws with 0xAA and never re-poisons between
// timed replays, so we must clear our accumulators every call).
__global__ void zero_kernel(float* __restrict__ p, long n) {
  long i = (long)blockIdx.x * blockDim.x + threadIdx.x;
  long stride = (long)gridDim.x * blockDim.x;
  for (; i < n; i += stride) p[i] = 0.0f;
}

// ---------------------------------------------------------------------------
// Weighted degrees: deg_out[src] += w, deg_in[dst] += w  (L2-resident atomics)
__global__ void degree_kernel(const int* __restrict__ ei, const float* __restrict__ ew,
                              float* __restrict__ deg_out, float* __restrict__ deg_in, int E) {
  int e = blockIdx.x * blockDim.x + threadIdx.x;
  if (e >= E) return;
  int s = ei[e];
  int d = ei[E + e];
  float w = ew[e];
  atomicAdd(&deg_out[s], w);
  atomicAdd(&deg_in[d], w);
}

// In-place reciprocal (nodes appearing as src/dst always have deg > 0).
__global__ void invert_kernel(float* __restrict__ p, long n) {
  long i = (long)blockIdx.x * blockDim.x + threadIdx.x;
  if (i < n) p[i] = 1.0f / p[i];
}

// ---------------------------------------------------------------------------
// One diffusion hop: tout[sidx[e]][:] += invdeg[gidx[e]] * tin[gidx[e]][:]
// 8 threads per edge, each moving a float4 chunk of the 32 channels.
__global__ void prop_kernel(const int* __restrict__ gidx, const int* __restrict__ sidx,
                            const float* __restrict__ invdeg, const float* __restrict__ tin,
                            float* __restrict__ tout, int E) {
  long t = (long)blockIdx.x * blockDim.x + threadIdx.x;
  if (t >= (long)E * 8) return;
  int e = (int)(t >> 3);
  int c = (int)(t & 7) << 2;
  int g = gidx[e];
  int sc = sidx[e];
  float coeff = invdeg[g];
  const float4 v = *reinterpret_cast<const float4*>(tin + (long)g * 32 + c);
  float* dst = tout + (long)sc * 32 + c;
  atomicAdd(dst + 0, v.x * coeff);
  atomicAdd(dst + 1, v.y * coeff);
  atomicAdd(dst + 2, v.z * coeff);
  atomicAdd(dst + 3, v.w * coeff);
}

// ---------------------------------------------------------------------------
// Fused GRU-gate + output-projection kernel, 1 wave (32 lanes) per 16 nodes.
// V_WMMA_F32_16X16X4_F32:  D(16x16,f32) += A(16x4,f32) x B(4x16,f32)
//   A frag (per ISA): lane l -> row = l&15, K pair = (l>>4)*2  -> v2f
//   C/D frag:         lane l -> col = l&15, rows = vgpr j + (l>=16 ? 8 : 0)
__global__ __launch_bounds__(32) void gate_kernel(
    const float* __restrict__ x,
    const float* __restrict__ to1, const float* __restrict__ to2,
    const float* __restrict__ ti1, const float* __restrict__ ti2,
    const float* __restrict__ wz, const float* __restrict__ bz,
    const float* __restrict__ wh, const float* __restrict__ bh,
    const float* __restrict__ lw, const float* __restrict__ lb,
    float* __restrict__ out) {
  __shared__ float lds[16 * 64];

  const int lane  = threadIdx.x;
  const int col   = lane & 15;          // A row index / B,C,D column index
  const int kb    = (lane >> 4) << 1;   // K sub-offset for A/B fragments
  const int rbase = (lane >> 4) << 3;   // C/D row base
  const long node0 = (long)blockIdx.x * 16;

  // Preload A fragments for the 5 hop sources: [src][kstep] -> float2
  const float* srcs[5] = { x, to1, to2, ti1, ti2 };
  v2f a[5][8];
  const long arow = (node0 + col) * 32;
#pragma unroll
  for (int h = 0; h < 5; ++h) {
#pragma unroll
    for (int s = 0; s < 8; ++s) {
      const float* p = srcs[h] + arow + s * 4 + kb;
      a[h][s].x = p[0];
      a[h][s].y = p[1];
    }
  }

  // 6 GEMM terms: (A source, weight slice index d*3+k) — H0==0 kills cols 32..95
  const int tsrc[6] = { 0, 0, 1, 2, 3, 4 };       // x, x, To1, To2, Ti1, Ti2
  const int tdk[6]  = { 0, 3, 1, 2, 4, 5 };       // (0,0),(1,0),(0,1),(0,2),(1,1),(1,2)

#pragma unroll
  for (int nt = 0; nt < 4; ++nt) {                // 4 tiles of 16 output channels
    v8f zacc = {};
    v8f hacc = {};
#pragma unroll
    for (int t = 0; t < 6; ++t) {
      const float* wzb = wz + (long)tdk[t] * 96 * 64 + nt * 16 + col;
      const float* whb = wh + (long)tdk[t] * 96 * 64 + nt * 16 + col;
#pragma unroll
      for (int s = 0; s < 8; ++s) {
        const int k0 = s * 4 + kb;
        v2f bzv; bzv.x = wzb[(long)k0 * 64]; bzv.y = wzb[(long)(k0 + 1) * 64];
        zacc = __builtin_amdgcn_wmma_f32_16x16x4_f32(
            false, a[tsrc[t]][s], false, bzv, (short)0, zacc, false, false);
        v2f bhv; bhv.x = whb[(long)k0 * 64]; bhv.y = whb[(long)(k0 + 1) * 64];
        hacc = __builtin_amdgcn_wmma_f32_16x16x4_f32(
            false, a[tsrc[t]][s], false, bhv, (short)0, hacc, false, false);
      }
    }
    const float biasz = bz[nt * 16 + col];
    const float biash = bh[nt * 16 + col];
#pragma unroll
    for (int j = 0; j < 8; ++j) {
      float z  = 1.0f / (1.0f + expf(-(zacc[j] + biasz)));
      float ht = tanhf(hacc[j] + biash);
      float hs = (1.0f - z) * ht;          // Z*H0 + (1-Z)*H_tilde with H0==0
      lds[(rbase + j) * 64 + nt * 16 + col] = fmaxf(hs, 0.0f);  // relu
    }
  }
  __syncthreads();

  // out(16x32) = relu(Hs)(16x64) @ lin_w(64x32) + lin_b
#pragma unroll
  for (int nt = 0; nt < 2; ++nt) {
    v8f acc = {};
#pragma unroll
    for (int s = 0; s < 16; ++s) {
      const int k0 = s * 4 + kb;
      v2f av; av.x = lds[col * 64 + k0]; av.y = lds[col * 64 + k0 + 1];
      v2f bv; bv.x = lw[(long)k0 * 32 + nt * 16 + col];
              bv.y = lw[(long)(k0 + 1) * 32 + nt * 16 + col];
      acc = __builtin_amdgcn_wmma_f32_16x16x4_f32(
          false, av, false, bv, (short)0, acc, false, false);
    }
    const float bb = lb[nt * 16 + col];
#pragma unroll
    for (int j = 0; j < 8; ++j) {
      out[(node0 + rbase + j) * 32 + nt * 16 + col] = acc[j] + bb;
    }
  }
}

// ---------------------------------------------------------------------------
extern "C" void kernel_launch(void* const* d_in, const int* in_sizes, int n_in,
                              void* d_out, int out_size, void* d_ws, size_t ws_size,
                              hipStream_t stream) {
  const float* x  = (const float*)d_in[0];
  const int*   ei = (const int*)d_in[1];
  const float* ew = (const float*)d_in[2];
  const float* wz = (const float*)d_in[3];
  const float* bz = (const float*)d_in[4];
  // d_in[5] (w_r) and d_in[6] (b_r) are mathematically dead: R * H0 == 0.
  const float* wh = (const float*)d_in[7];
  const float* bh = (const float*)d_in[8];
  const float* lw = (const float*)d_in[9];
  const float* lb = (const float*)d_in[10];
  float* out = (float*)d_out;
  float* ws  = (float*)d_ws;

  const long N = NN;
  const int  E = NE;

  float* dego = ws;                 // N   -> becomes 1/deg_out
  float* degi = ws + N;             // N   -> becomes 1/deg_in
  float* To1  = ws + 2 * N;         // 32N
  float* Ti1  = To1 + 32 * N;       // 32N
  float* To2  = Ti1 + 32 * N;       // 32N
  float* Ti2  = To2 + 32 * N;       // 32N  (total 130N floats = 52 MB)

  const long zcount = 2 * N + 4 * 32 * N;
  zero_kernel<<<2048, 256, 0, stream>>>(ws, zcount);

  degree_kernel<<<(E + 255) / 256, 256, 0, stream>>>(ei, ew, dego, degi, E);
  invert_kernel<<<(int)((2 * N + 255) / 256), 256, 0, stream>>>(ws, 2 * N);

  const long pth = (long)E * 8;
  const int pblocks = (int)((pth + 255) / 256);
  // forward walk hops: gather src, scale by 1/deg_out(src), scatter to dst
  prop_kernel<<<pblocks, 256, 0, stream>>>(ei,     ei + E, dego, x,   To1, E);
  // reverse walk hops: gather dst, scale by 1/deg_in(dst), scatter to src
  prop_kernel<<<pblocks, 256, 0, stream>>>(ei + E, ei,     degi, x,   Ti1, E);
  prop_kernel<<<pblocks, 256, 0, stream>>>(ei,     ei + E, dego, To1, To2, E);
  prop_kernel<<<pblocks, 256, 0, stream>>>(ei + E, ei,     degi, Ti1, Ti2, E);

  gate_kernel<<<NN / 16, 32, 0, stream>>>(x, To1, To2, Ti1, Ti2,
                                          wz, bz, wh, bh, lw, lb, out);
}